// MultiPointExtractor_35545149342110
// MI455X (gfx1250) — compile-verified
//
#include <hip/hip_runtime.h>
#include <cstdint>

typedef __attribute__((ext_vector_type(4))) unsigned int u32x4;
typedef __attribute__((ext_vector_type(8))) unsigned int u32x8;

#define NROI 512
#define NPTS 128
#define CPL  256    // channels per level

// grid.x = NROI * 4 levels * 4 channel-quarters = 8192
// block  = 256 threads = 128 points x 2 channel-halves (each half: 32 channels)
__global__ __launch_bounds__(256) void
mpe_sample_kernel(const float* __restrict__ f0, const float* __restrict__ f1,
                  const float* __restrict__ f2, const float* __restrict__ f3,
                  const float* __restrict__ rois, const float* __restrict__ points,
                  float* __restrict__ out)
{
    const int bx  = blockIdx.x;
    const int n   = bx >> 4;          // roi
    const int lev = (bx >> 2) & 3;    // pyramid level
    const int cq  = bx & 3;           // channel quarter (64 channels)
    const int tid = threadIdx.x;
    const int p   = tid & (NPTS - 1);
    const int cbase = cq * 64 + (tid >> 7) * 32;

    __shared__ float s_pts[2 * NPTS];

    // --- TDM: DMA this ROI's 128 (x,y) points (1 KB contiguous) into LDS ---
    // 2-SGPR-group descriptor form (VADDR2/VADDR3 = NULL): valid for <=2D tiles.
    if (tid < 32) {   // wave 0 only (TDM issues per-wave, EXEC ignored)
        const uint32_t lds_off = (uint32_t)(uintptr_t)(void*)s_pts;
        const uint64_t ga = (uint64_t)(uintptr_t)(points + (size_t)n * (2 * NPTS));
        u32x4 g0;
        g0.x = 1u;                                                // count=1 (valid user D#)
        g0.y = lds_off;                                           // lds_addr (bytes)
        g0.z = (uint32_t)ga;                                      // global_addr[31:0]
        g0.w = (uint32_t)((ga >> 32) & 0x01FFFFFFu) | (2u << 30); // addr[56:32] | type=2
        u32x8 g1;
        g1[0] = 0x00020000u;      // workgroup_mask=0, data_size=2 (4 bytes)
        g1[1] = 256u << 16;       // tensor_dim0[15:0]=256 in bits[63:48]
        g1[2] = 1u << 16;         // tensor_dim0.hi=0 | tensor_dim1[15:0]=1
        g1[3] = 256u << 16;       // tensor_dim1.hi=0 | tile_dim0=256
        g1[4] = 1u;               // tile_dim1=1, tile_dim2=0 (unused)
        g1[5] = 256u;             // tensor_dim0_stride[31:0]=256
        g1[6] = 0u;               // stride0.hi=0 | tensor_dim1_stride.lo=0
        g1[7] = 0u;               // tensor_dim1_stride.hi=0 (row count is 1)
        asm volatile("tensor_load_to_lds %0, %1"
                     :
                     : "s"(g0), "s"(g1)
                     : "memory");
        __builtin_amdgcn_s_wait_tensorcnt(0);
    }
    __syncthreads();

    // --- per-point bilinear setup (computed once, reused for 32 channels) ---
    const float* r = rois + (size_t)n * 5;
    const int   b  = (int)r[0];
    const float x1 = r[1], y1 = r[2];
    const float dx = r[3] - x1, dy = r[4] - y1;

    const float ptx = s_pts[2 * p];
    const float pty = s_pts[2 * p + 1];
    const float cx = x1 + ptx * dx;
    const float cy = y1 + pty * dy;

    const int   Wd = 256 >> lev;
    const int   HW = Wd * Wd;
    // exact 1/(4<<lev): 0.25f with exponent reduced by lev
    const float inv_stride = __int_as_float(0x3E800000 - (lev << 23));

    const float px = cx * inv_stride - 0.5f;
    const float py = cy * inv_stride - 0.5f;
    const float x0f = floorf(px), y0f = floorf(py);
    const float wx = px - x0f,    wy = py - y0f;
    const int x0 = (int)x0f, y0 = (int)y0f;

    const bool vx0 = (x0 >= 0)     && (x0 < Wd);
    const bool vx1 = (x0 + 1 >= 0) && (x0 + 1 < Wd);
    const bool vy0 = (y0 >= 0)     && (y0 < Wd);
    const bool vy1 = (y0 + 1 >= 0) && (y0 + 1 < Wd);
    const int xc0 = min(max(x0, 0),     Wd - 1);
    const int xc1 = min(max(x0 + 1, 0), Wd - 1);
    const int yc0 = min(max(y0, 0),     Wd - 1);
    const int yc1 = min(max(y0 + 1, 0), Wd - 1);

    const int o00 = yc0 * Wd + xc0;
    const int o01 = yc0 * Wd + xc1;
    const int o10 = yc1 * Wd + xc0;
    const int o11 = yc1 * Wd + xc1;

    // fold validity into weights (identical to reference's v * valid)
    float w00 = (1.0f - wx) * (1.0f - wy); if (!(vx0 && vy0)) w00 = 0.0f;
    float w01 = wx * (1.0f - wy);          if (!(vx1 && vy0)) w01 = 0.0f;
    float w10 = (1.0f - wx) * wy;          if (!(vx0 && vy1)) w10 = 0.0f;
    float w11 = wx * wy;                   if (!(vx1 && vy1)) w11 = 0.0f;

    const float* fb = (lev == 0) ? f0 : (lev == 1) ? f1 : (lev == 2) ? f2 : f3;
    const float* f  = fb + ((size_t)b * CPL + (size_t)cbase) * (size_t)HW;
    float* ob = out + (((size_t)n * (4 * CPL)) + (size_t)lev * CPL + (size_t)cbase) * NPTS + p;

    #pragma unroll 8
    for (int c = 0; c < 32; ++c) {
        // warm the two cachelines (top/bottom tap rows) 4 channel planes ahead
        __builtin_prefetch(f + (size_t)4 * HW + o00, 0, 0);
        __builtin_prefetch(f + (size_t)4 * HW + o10, 0, 0);
        float v = w00 * f[o00];
        v = fmaf(w01, f[o01], v);
        v = fmaf(w10, f[o10], v);
        v = fmaf(w11, f[o11], v);
        // non-temporal: output is write-once (256 MB); keep L2 for the
        // 178 MB feature working set that the gathers re-traverse.
        __builtin_nontemporal_store(v, ob);
        f  += HW;       // next channel plane (scalar add)
        ob += NPTS;     // next channel row of output
    }
}

extern "C" void kernel_launch(void* const* d_in, const int* in_sizes, int n_in,
                              void* d_out, int out_size, void* d_ws, size_t ws_size,
                              hipStream_t stream) {
    const float* f0     = (const float*)d_in[0];
    const float* f1     = (const float*)d_in[1];
    const float* f2     = (const float*)d_in[2];
    const float* f3     = (const float*)d_in[3];
    const float* rois   = (const float*)d_in[4];
    const float* points = (const float*)d_in[5];
    float* out = (float*)d_out;

    dim3 grid(NROI * 4 * 4);
    dim3 block(256);
    hipLaunchKernelGGL(mpe_sample_kernel, grid, block, 0, stream,
                       f0, f1, f2, f3, rois, points, out);
}